// HPGMambaStage_43602507989587
// MI455X (gfx1250) — compile-verified
//
#include <hip/hip_runtime.h>
#include <hip/hip_bf16.h>
#include <math.h>

// ---------------------------------------------------------------------------
// HPGMamba stage for MI455X (gfx1250, wave32).
// All five GEMMs run on V_WMMA_F32_16X16X4_F32 (exact fp32 matrix path).
// Heavy mixing GEMMs (u,y: ~17 GFLOP each) stage their shared B-operand slabs in
// LDS (64KB, bank-conflict-padded) so 8 waves/block reuse one cooperative load,
// via GLOBAL_LOAD_ASYNC_TO_LDS_B128 (ASYNCcnt) when the builtin is available.
// Working set (~130MB) fits in 192MB L2, so remaining global traffic is L2-hit.
// ---------------------------------------------------------------------------

typedef __attribute__((ext_vector_type(2))) float v2f;
typedef __attribute__((ext_vector_type(8))) float v8f;
typedef __attribute__((ext_vector_type(4))) int   v4i;

#define BD  4
#define CIN 256
#define LL  4096
#define DD  256
#define NN  512
#define KK  4
#define DMID 128
#define LDSTRIDE 260   // 256 + 4 pad: lane m -> bank (4*m + kd) % 64, conflict-free

#if defined(__has_builtin)
#if __has_builtin(__builtin_amdgcn_global_load_async_to_lds_b128) && \
    __has_builtin(__builtin_amdgcn_s_wait_asynccnt)
#define USE_ASYNC_LDS 1
#endif
#endif

__device__ __forceinline__ v8f wmma4(v2f a, v2f b, v8f c) {
  // D = A(16x4 f32) * B(4x16 f32) + C(16x16 f32)
  return __builtin_amdgcn_wmma_f32_16x16x4_f32(false, a, false, b,
                                               (short)0, c, false, false);
}

// Cooperative stage of a 64-row x 256-col f32 slab into padded LDS.
__device__ __forceinline__ void stage64x256(float* Bs, const float* src,
                                            int rowStride) {
  const int tid = threadIdx.x;
#if USE_ASYNC_LDS
#pragma unroll 4
  for (int i = tid; i < 64 * 64; i += 256) {
    const int row = i >> 6, c4 = (i & 63) << 2;
    __builtin_amdgcn_global_load_async_to_lds_b128(
        (v4i*)(src + (size_t)row * rowStride + c4),
        (v4i*)(Bs + row * LDSTRIDE + c4),
        0, 0);
  }
  __builtin_amdgcn_s_wait_asynccnt(0);
#else
#pragma unroll 4
  for (int i = tid; i < 64 * 64; i += 256) {
    const int row = i >> 6, c4 = (i & 63) << 2;
    *(float4*)(Bs + row * LDSTRIDE + c4) =
        *(const float4*)(src + (size_t)row * rowStride + c4);
  }
#endif
}

// ---------------------------------------------------------------------------
// Kernel 1: dual 1x1-conv projection.
//   x[b,l,d]   = sum_c feat[b,c,l]*Wf[d,c]  + bf[d]
//   thf[b,l,d] = sum_c hf[b,c,l]  *Whf[d,c] + bhf[d]
// One wave = 16(l) x 32(d) tile for BOTH streams (4 accumulators).
// ---------------------------------------------------------------------------
__global__ __launch_bounds__(256) void k_proj(
    const float* __restrict__ feat, const float* __restrict__ hf,
    const float* __restrict__ Wf,  const float* __restrict__ bf,
    const float* __restrict__ Whf, const float* __restrict__ bhf,
    float* __restrict__ x, float* __restrict__ thf) {
  const int lane = threadIdx.x & 31;
  const int wave = (blockIdx.x * blockDim.x + threadIdx.x) >> 5;
  const int DT = DD / 32;                       // 8 d-groups of 32
  const int b   = wave / ((LL / 16) * DT);
  const int rem = wave % ((LL / 16) * DT);
  const int l0  = (rem / DT) * 16;
  const int d0  = (rem % DT) * 32;
  const int m = lane & 15, kh = lane >> 4;

  const float* fb = feat + (size_t)b * CIN * LL;
  const float* hb = hf   + (size_t)b * CIN * LL;

  v8f af0 = (v8f)0.0f, af1 = (v8f)0.0f, ah0 = (v8f)0.0f, ah1 = (v8f)0.0f;
  for (int c = 0; c < CIN; c += 4) {
    const int kc = c + 2 * kh;
    v2f a_f, a_h;                               // A frag: A[l][c]=feat[b][c][l]
    a_f.x = fb[(size_t)kc * LL + l0 + m];
    a_f.y = fb[(size_t)(kc + 1) * LL + l0 + m];
    a_h.x = hb[(size_t)kc * LL + l0 + m];
    a_h.y = hb[(size_t)(kc + 1) * LL + l0 + m];
    // B frag: B[c][d] = W[d*CIN + c]; (kc,kc+1) contiguous -> float2
    v2f bf0 = *(const v2f*)(Wf  + (size_t)(d0 + m)      * CIN + kc);
    v2f bf1 = *(const v2f*)(Wf  + (size_t)(d0 + 16 + m) * CIN + kc);
    v2f bh0 = *(const v2f*)(Whf + (size_t)(d0 + m)      * CIN + kc);
    v2f bh1 = *(const v2f*)(Whf + (size_t)(d0 + 16 + m) * CIN + kc);
    af0 = wmma4(a_f, bf0, af0);
    af1 = wmma4(a_f, bf1, af1);
    ah0 = wmma4(a_h, bh0, ah0);
    ah1 = wmma4(a_h, bh1, ah1);
  }
  const float b0 = bf[d0 + m],  b1 = bf[d0 + 16 + m];
  const float c0 = bhf[d0 + m], c1 = bhf[d0 + 16 + m];
  float* xo = x   + ((size_t)b * LL + l0) * DD;
  float* ho = thf + ((size_t)b * LL + l0) * DD;
#pragma unroll
  for (int r = 0; r < 8; r++) {
    const int row = r + kh * 8;                 // C/D frag: row=r(+8), col=m
    xo[(size_t)row * DD + d0 + m]      = af0[r] + b0;
    xo[(size_t)row * DD + d0 + 16 + m] = af1[r] + b1;
    ho[(size_t)row * DD + d0 + m]      = ah0[r] + c0;
    ho[(size_t)row * DD + d0 + 16 + m] = ah1[r] + c1;
  }
}

// ---------------------------------------------------------------------------
// Kernel 2: gating head. One wave per token: 32 lanes x 4 mid-units each,
// exact GELU (erf), butterfly-reduced logits, softmax over K=4.
// ---------------------------------------------------------------------------
__global__ __launch_bounds__(256) void k_alpha(
    const float* __restrict__ thf,
    const float* __restrict__ Wa1, const float* __restrict__ ba1,
    const float* __restrict__ Wa2, const float* __restrict__ ba2,
    float* __restrict__ alpha) {
  const int lane = threadIdx.x & 31;
  const int t = (blockIdx.x * blockDim.x + threadIdx.x) >> 5;  // token in [0,B*L)
  const float* tp = thf + (size_t)t * DD;

  float mid[4];
#pragma unroll
  for (int jj = 0; jj < 4; jj++) {
    const int j = lane + jj * 32;
    const float* w = Wa1 + (size_t)j * DD;
    float s = ba1[j];
    for (int d = 0; d < DD; d += 4) {
      const float4 tv = *(const float4*)(tp + d);
      const float4 wv = *(const float4*)(w + d);
      s = fmaf(tv.x, wv.x, s); s = fmaf(tv.y, wv.y, s);
      s = fmaf(tv.z, wv.z, s); s = fmaf(tv.w, wv.w, s);
    }
    mid[jj] = 0.5f * s * (1.0f + erff(s * 0.70710678118654752f));  // exact GELU
  }
  float logit[KK];
#pragma unroll
  for (int kk = 0; kk < KK; kk++) {
    float p = 0.0f;
#pragma unroll
    for (int jj = 0; jj < 4; jj++)
      p = fmaf(mid[jj], Wa2[kk * DMID + lane + jj * 32], p);
#pragma unroll
    for (int off = 16; off >= 1; off >>= 1) p += __shfl_xor(p, off, 32);
    logit[kk] = p + ba2[kk];
  }
  if (lane == 0) {
    const float mx = fmaxf(fmaxf(logit[0], logit[1]), fmaxf(logit[2], logit[3]));
    float e[KK]; float sum = 0.0f;
#pragma unroll
    for (int kk = 0; kk < KK; kk++) { e[kk] = expf(logit[kk] - mx); sum += e[kk]; }
    const float inv = 1.0f / sum;
#pragma unroll
    for (int kk = 0; kk < KK; kk++) alpha[(size_t)t * KK + kk] = e[kk] * inv;
  }
}

// ---------------------------------------------------------------------------
// Kernel 3: u[b,l,n] = sum_k alpha[l,k] * (x[l,:] . B_base[k,n,:])
//           a[b,l,n] = sigmoid(sum_k alpha[l,k]*A_base[k,n])
// Block = (b, 256 l-rows, 64 n-cols): 8 waves x 32 l-rows each. Per k, the
// 64x256 B_base slab is staged once in LDS and reused by all 8 waves.
// alpha is folded into the A-fragment rows -> single accumulator set over k.
// ---------------------------------------------------------------------------
__global__ __launch_bounds__(256) void k_mixu(
    const float* __restrict__ x, const float* __restrict__ alpha,
    const float* __restrict__ Bb, const float* __restrict__ Ab,
    float* __restrict__ u, float* __restrict__ adec) {
  __shared__ float Bs[64 * LDSTRIDE];
  const int lane = threadIdx.x & 31;
  const int w    = threadIdx.x >> 5;
  const int NB = NN / 64, LS = LL / 256;        // 8, 16
  const int b   = blockIdx.x / (LS * NB);
  const int rem = blockIdx.x % (LS * NB);
  const int l0  = (rem / NB) * 256 + w * 32;    // this wave: rows l0..l0+31
  const int n0  = (rem % NB) * 64;
  const int m = lane & 15, kh = lane >> 4;

  const float* xp0 = x + ((size_t)b * LL + l0 + m) * DD;
  const float* xp1 = xp0 + (size_t)16 * DD;
  const float4 av0 = *(const float4*)(alpha + ((size_t)b * LL + l0 + m) * KK);
  const float4 av1 = *(const float4*)(alpha + ((size_t)b * LL + l0 + 16 + m) * KK);
  const float alf0[KK] = {av0.x, av0.y, av0.z, av0.w};
  const float alf1[KK] = {av1.x, av1.y, av1.z, av1.w};

  v8f acc0[4], acc1[4];
#pragma unroll
  for (int i = 0; i < 4; i++) { acc0[i] = (v8f)0.0f; acc1[i] = (v8f)0.0f; }

  for (int k = 0; k < KK; k++) {
    __syncthreads();                            // previous slab consumed
    stage64x256(Bs, Bb + (size_t)k * NN * DD + (size_t)n0 * DD, DD);
    __syncthreads();                            // slab visible to all waves
    const float ak0 = alf0[k], ak1 = alf1[k];
    for (int d = 0; d < DD; d += 4) {
      const int kd = d + 2 * kh;
      v2f a0 = *(const v2f*)(xp0 + kd); a0.x *= ak0; a0.y *= ak0;
      v2f a1 = *(const v2f*)(xp1 + kd); a1.x *= ak1; a1.y *= ak1;
#pragma unroll
      for (int nt = 0; nt < 4; nt++) {
        const v2f bfr = *(const v2f*)(Bs + (nt * 16 + m) * LDSTRIDE + kd);
        acc0[nt] = wmma4(a0, bfr, acc0[nt]);    // rows l0..l0+15
        acc1[nt] = wmma4(a1, bfr, acc1[nt]);    // rows l0+16..l0+31
      }
    }
  }
  float* up = u    + ((size_t)b * LL + l0) * NN;
  float* ap = adec + ((size_t)b * LL + l0) * NN;
#pragma unroll
  for (int g = 0; g < 2; g++) {
#pragma unroll
    for (int r = 0; r < 8; r++) {
      const int row = g * 16 + r + kh * 8;
      const float4 ar = *(const float4*)(alpha + ((size_t)b * LL + l0 + row) * KK);
#pragma unroll
      for (int nt = 0; nt < 4; nt++) {
        const int n = n0 + nt * 16 + m;
        up[(size_t)row * NN + n] = (g ? acc1[nt][r] : acc0[nt][r]);
        const float s = ar.x * Ab[0 * NN + n] + ar.y * Ab[1 * NN + n] +
                        ar.z * Ab[2 * NN + n] + ar.w * Ab[3 * NN + n];
        ap[(size_t)row * NN + n] = 1.0f / (1.0f + expf(-s));
      }
    }
  }
}

// ---------------------------------------------------------------------------
// Kernel 4: selective scan h[l] = a[l]*h[l-1] + u[l], in-place over u.
// One lane per (b,n): lane index == n -> fully coalesced 128B wave accesses.
// ---------------------------------------------------------------------------
__global__ __launch_bounds__(256) void k_scan(const float* __restrict__ adec,
                                              float* __restrict__ u) {
  const int tid = blockIdx.x * blockDim.x + threadIdx.x;  // [0, B*N)
  const int b = tid / NN, n = tid % NN;
  const float* ap = adec + (size_t)b * LL * NN + n;
  float* up = u + (size_t)b * LL * NN + n;
  float h = 0.0f;
  for (int l = 0; l < LL; l += 4) {
    __builtin_prefetch(ap + (size_t)(l + 16) * NN, 0, 0);  // global_prefetch_b8
    __builtin_prefetch(up + (size_t)(l + 16) * NN, 0, 0);
    const float a0 = ap[(size_t)(l + 0) * NN], u0 = up[(size_t)(l + 0) * NN];
    const float a1 = ap[(size_t)(l + 1) * NN], u1 = up[(size_t)(l + 1) * NN];
    const float a2 = ap[(size_t)(l + 2) * NN], u2 = up[(size_t)(l + 2) * NN];
    const float a3 = ap[(size_t)(l + 3) * NN], u3 = up[(size_t)(l + 3) * NN];
    h = fmaf(a0, h, u0); up[(size_t)(l + 0) * NN] = h;
    h = fmaf(a1, h, u1); up[(size_t)(l + 1) * NN] = h;
    h = fmaf(a2, h, u2); up[(size_t)(l + 2) * NN] = h;
    h = fmaf(a3, h, u3); up[(size_t)(l + 3) * NN] = h;
  }
}

// ---------------------------------------------------------------------------
// Kernel 5: y[b,l,d] = sum_k alpha[l,k] * (h[l,:] . C_base[k,d,:])
// Block = (b, 256 l-rows, 64 d-cols). K-dim (n) is 512 -> LDS-stage the
// 64x512 C_base slab in two 64x256 chunks per k.
// ---------------------------------------------------------------------------
__global__ __launch_bounds__(256) void k_mixy(
    const float* __restrict__ h, const float* __restrict__ alpha,
    const float* __restrict__ Cb, float* __restrict__ y) {
  __shared__ float Cs[64 * LDSTRIDE];
  const int lane = threadIdx.x & 31;
  const int w    = threadIdx.x >> 5;
  const int DB = DD / 64, LS = LL / 256;        // 4, 16
  const int b   = blockIdx.x / (LS * DB);
  const int rem = blockIdx.x % (LS * DB);
  const int l0  = (rem / DB) * 256 + w * 32;
  const int d0  = (rem % DB) * 64;
  const int m = lane & 15, kh = lane >> 4;

  const float* hp0 = h + ((size_t)b * LL + l0 + m) * NN;
  const float* hp1 = hp0 + (size_t)16 * NN;
  const float4 av0 = *(const float4*)(alpha + ((size_t)b * LL + l0 + m) * KK);
  const float4 av1 = *(const float4*)(alpha + ((size_t)b * LL + l0 + 16 + m) * KK);
  const float alf0[KK] = {av0.x, av0.y, av0.z, av0.w};
  const float alf1[KK] = {av1.x, av1.y, av1.z, av1.w};

  v8f acc0[4], acc1[4];
#pragma unroll
  for (int i = 0; i < 4; i++) { acc0[i] = (v8f)0.0f; acc1[i] = (v8f)0.0f; }

  for (int k = 0; k < KK; k++) {
    const float ak0 = alf0[k], ak1 = alf1[k];
    for (int ch = 0; ch < 2; ch++) {            // two 256-wide n chunks
      __syncthreads();
      stage64x256(Cs, Cb + (size_t)k * DD * NN + (size_t)d0 * NN + ch * 256, NN);
      __syncthreads();
      const int nbase = ch * 256;
      for (int nn = 0; nn < 256; nn += 4) {
        const int kn = nbase + nn + 2 * kh;
        v2f a0 = *(const v2f*)(hp0 + kn); a0.x *= ak0; a0.y *= ak0;
        v2f a1 = *(const v2f*)(hp1 + kn); a1.x *= ak1; a1.y *= ak1;
#pragma unroll
        for (int nt = 0; nt < 4; nt++) {
          const v2f bfr = *(const v2f*)(Cs + (nt * 16 + m) * LDSTRIDE + nn + 2 * kh);
          acc0[nt] = wmma4(a0, bfr, acc0[nt]);
          acc1[nt] = wmma4(a1, bfr, acc1[nt]);
        }
      }
    }
  }
  float* yp = y + ((size_t)b * LL + l0) * DD;
#pragma unroll
  for (int g = 0; g < 2; g++) {
#pragma unroll
    for (int r = 0; r < 8; r++) {
      const int row = g * 16 + r + kh * 8;
#pragma unroll
      for (int nt = 0; nt < 4; nt++)
        yp[(size_t)row * DD + d0 + nt * 16 + m] = (g ? acc1[nt][r] : acc0[nt][r]);
    }
  }
}

// ---------------------------------------------------------------------------
// Kernel 6: out[b,c,l] = sum_d y[b,l,d]*Wout[c,d] + bout[c]
// M=c, N=l (4-way tiled), K=d. A frag (Wout, 256KB weights, L2/L0-hot).
// ---------------------------------------------------------------------------
__global__ __launch_bounds__(256) void k_out(
    const float* __restrict__ y, const float* __restrict__ Wout,
    const float* __restrict__ bout, float* __restrict__ out) {
  const int lane = threadIdx.x & 31;
  const int wave = (blockIdx.x * blockDim.x + threadIdx.x) >> 5;
  const int LB = LL / 64;                       // 64
  const int b   = wave / ((CIN / 16) * LB);
  const int rem = wave % ((CIN / 16) * LB);
  const int c0  = (rem / LB) * 16;
  const int l0  = (rem % LB) * 64;
  const int m = lane & 15, kh = lane >> 4;

  const float* yb = y + (size_t)b * LL * DD;
  v8f acc[4];
#pragma unroll
  for (int i = 0; i < 4; i++) acc[i] = (v8f)0.0f;

  for (int d = 0; d < DD; d += 4) {
    const int kd = d + 2 * kh;
    const v2f a = *(const v2f*)(Wout + (size_t)(c0 + m) * DD + kd);
#pragma unroll
    for (int nt = 0; nt < 4; nt++) {
      // B[d][l] = y[l*DD + d]; (kd,kd+1) contiguous -> float2
      const v2f bfr = *(const v2f*)(yb + (size_t)(l0 + nt * 16 + m) * DD + kd);
      acc[nt] = wmma4(a, bfr, acc[nt]);
    }
  }
  float* op = out + ((size_t)b * CIN + c0) * LL;
#pragma unroll
  for (int r = 0; r < 8; r++) {
    const int row = r + kh * 8;
    const float bb = bout[c0 + row];
#pragma unroll
    for (int nt = 0; nt < 4; nt++)
      op[(size_t)row * LL + l0 + nt * 16 + m] = acc[nt][r] + bb;
  }
}

// ---------------------------------------------------------------------------
extern "C" void kernel_launch(void* const* d_in, const int* in_sizes, int n_in,
                              void* d_out, int out_size, void* d_ws, size_t ws_size,
                              hipStream_t stream) {
  const float* feat = (const float*)d_in[0];
  const float* hf   = (const float*)d_in[1];
  const float* Wf   = (const float*)d_in[2];
  const float* bf   = (const float*)d_in[3];
  const float* Whf  = (const float*)d_in[4];
  const float* bhf  = (const float*)d_in[5];
  const float* Wa1  = (const float*)d_in[6];
  const float* ba1  = (const float*)d_in[7];
  const float* Wa2  = (const float*)d_in[8];
  const float* ba2  = (const float*)d_in[9];
  const float* Ab   = (const float*)d_in[10];
  const float* Bb   = (const float*)d_in[11];
  const float* Cb   = (const float*)d_in[12];
  const float* Wout = (const float*)d_in[13];
  const float* bout = (const float*)d_in[14];
  float* out = (float*)d_out;

  // Workspace layout (floats): x | thf(->y) | alpha | a | u(->h)  ~101 MB
  float* x     = (float*)d_ws;
  float* thf   = x     + (size_t)BD * LL * DD;   // also reused as y
  float* alpha = thf   + (size_t)BD * LL * DD;
  float* adec  = alpha + (size_t)BD * LL * KK;
  float* u     = adec  + (size_t)BD * LL * NN;   // scan overwrites u with h

  // 1) projections: 8192 waves
  k_proj<<<1024, 256, 0, stream>>>(feat, hf, Wf, bf, Whf, bhf, x, thf);
  // 2) gating: one wave per token (16384 waves)
  k_alpha<<<2048, 256, 0, stream>>>(thf, Wa1, ba1, Wa2, ba2, alpha);
  // 3) u + decay: block = (b, 256 l, 64 n) with LDS slab reuse
  k_mixu<<<512, 256, 0, stream>>>(x, alpha, Bb, Ab, u, adec);
  // 4) scan: 2048 chains (lane==n, coalesced)
  k_scan<<<8, 256, 0, stream>>>(adec, u);
  // 5) readout mix: block = (b, 256 l, 64 d) with LDS slab reuse
  k_mixy<<<256, 256, 0, stream>>>(u, alpha, Cb, thf);
  // 6) output projection: 4096 waves
  k_out<<<512, 256, 0, stream>>>(thf, Wout, bout, out);
}